// EBSW_38242388804118
// MI455X (gfx1250) — compile-verified
//
#include <hip/hip_runtime.h>
#include <hip/hip_bf16.h>
#include <stdint.h>

#define B_ 64
#define N_ 4096
#define D_ 3
#define L_ 128

// LDS padding: +4 floats every 16 to soften bank conflicts on per-chunk B128 access
#define PADF(i) ((i) + (((i) >> 4) << 2))
#define NPAD (N_ + ((N_ >> 4) << 2))   // 5120

typedef __attribute__((ext_vector_type(2))) float v2f;
typedef __attribute__((ext_vector_type(8))) float v8f;

// ---------------- RNG (hash-based; JAX PRNG not reproducible, compile-only env) ---------

__device__ __forceinline__ uint32_t pcg_hash(uint32_t v) {
    v = v * 747796405u + 2891336453u;
    uint32_t w = ((v >> ((v >> 28u) + 4u)) ^ v) * 277803737u;
    return (w >> 22u) ^ w;
}
__device__ __forceinline__ float u01(uint32_t h) {
    return ((float)h + 0.5f) * 2.3283064365386963e-10f; // in (0,1)
}

// theta4: [B][L][4] unit directions (w = 0 pad), logu: [B][L] log-uniforms
__global__ void ebsw_rng(float* __restrict__ theta4, float* __restrict__ logu) {
    int i = blockIdx.x * blockDim.x + threadIdx.x; // over B*L
    if (i >= B_ * L_) return;
    uint32_t s = (uint32_t)i ^ 0x9E3779B9u;
    s = pcg_hash(s);
    float u1 = u01(s = pcg_hash(s));
    float u2 = u01(s = pcg_hash(s));
    float u3 = u01(s = pcg_hash(s));
    float u4 = u01(s = pcg_hash(s));
    float u5 = u01(s = pcg_hash(s));
    const float TWO_PI = 6.28318530717958647692f;
    float r1 = sqrtf(-2.0f * __logf(u1));
    float g0 = r1 * __cosf(TWO_PI * u2);
    float g1 = r1 * __sinf(TWO_PI * u2);
    float r2 = sqrtf(-2.0f * __logf(u3));
    float g2 = r2 * __cosf(TWO_PI * u4);
    float inv = rsqrtf(g0 * g0 + g1 * g1 + g2 * g2 + 1e-20f);
    theta4[(size_t)i * 4 + 0] = g0 * inv;
    theta4[(size_t)i * 4 + 1] = g1 * inv;
    theta4[(size_t)i * 4 + 2] = g2 * inv;
    theta4[(size_t)i * 4 + 3] = 0.0f;
    logu[i] = __logf(u5);
}

// ---------------- Projection via V_WMMA_F32_16X16X4_F32 -------------------------------
// Each wave: A = 16 points x 4 dims (dim3 padded 0), B = 4 x 16 directions,
// D = 16x16 tile of projections. 8 waves/block -> 128 points x 16 dirs per block.
__global__ void ebsw_proj_wmma(const float* __restrict__ x, const float* __restrict__ y,
                               const float* __restrict__ theta4,
                               float* __restrict__ xp, float* __restrict__ yp,
                               int l0, int Lcur) {
    const int lane = threadIdx.x & 31;
    const int wave = threadIdx.x >> 5;
    const int b    = blockIdx.z;
    const int lt   = blockIdx.y;                  // 16-direction tile within chunk
    const int p0   = blockIdx.x * 128 + wave * 16;
    const int n    = lane & 15;
    const bool lo  = lane < 16;

    // B matrix (4 x 16): lanes 0-15 hold K=0 (v0) / K=1 (v1); lanes 16-31 hold K=2 / pad
    const int lglob = l0 + lt * 16 + n;
    const float* tp = theta4 + ((size_t)b * L_ + lglob) * 4;
    v2f bm;
    bm.x = lo ? tp[0] : tp[2];
    bm.y = lo ? tp[1] : 0.0f;

    // A matrices (16 x 4) for x and y: lanes 0-15 hold dims 0,1; lanes 16-31 dims 2,pad
    const size_t pbase = ((size_t)b * N_ + p0 + n) * 3;
    v2f amx, amy;
    amx.x = lo ? x[pbase + 0] : x[pbase + 2];
    amx.y = lo ? x[pbase + 1] : 0.0f;
    amy.x = lo ? y[pbase + 0] : y[pbase + 2];
    amy.y = lo ? y[pbase + 1] : 0.0f;

    v8f c = {};
    v8f dx = __builtin_amdgcn_wmma_f32_16x16x4_f32(false, amx, false, bm, (short)0, c, false, false);
    v8f dy = __builtin_amdgcn_wmma_f32_16x16x4_f32(false, amy, false, bm, (short)0, c, false, false);

    // D layout: vgpr v, lanes 0-15 -> M=v, N=lane; lanes 16-31 -> M=v+8, N=lane-16
    const int lc   = lt * 16 + n;                 // direction index within chunk
    const int mofs = lo ? 0 : 8;
    const size_t rowbase = ((size_t)b * Lcur + lc) * N_ + p0 + mofs;
#pragma unroll
    for (int v = 0; v < 8; ++v) {
        xp[rowbase + v] = dx[v];
        yp[rowbase + v] = dy[v];
    }
}

// ---------------- Register-level bitonic helpers ---------------------------------------

__device__ __forceinline__ void cex(float& a, float& b, bool up) {
    float lo = fminf(a, b), hi = fmaxf(a, b);
    a = up ? lo : hi;
    b = up ? hi : lo;
}

template <int J>
__device__ __forceinline__ void reg_pass(float r[16], int base, unsigned k) {
#pragma unroll
    for (int v = 0; v < 16; ++v) {
        if ((v & J) == 0) {
            bool up = (((unsigned)(base + v) & k) == 0u);
            cex(r[v], r[v + J], up);
        }
    }
}

// strides jstart..1 (jstart in {1,2,4,8}, compile-time at call sites)
__device__ __forceinline__ void reg_tail(float rx[16], float ry[16], int base,
                                         unsigned k, int jstart) {
    if (jstart >= 8) { reg_pass<8>(rx, base, k); reg_pass<8>(ry, base, k); }
    if (jstart >= 4) { reg_pass<4>(rx, base, k); reg_pass<4>(ry, base, k); }
    if (jstart >= 2) { reg_pass<2>(rx, base, k); reg_pass<2>(ry, base, k); }
    reg_pass<1>(rx, base, k);
    reg_pass<1>(ry, base, k);
}

// ---------------- Hybrid LDS/register bitonic sort + squared-diff reduce ---------------
// Each thread owns 16 contiguous elements: strides j<=8 run in registers.
// LDS passes only for j>=16 -> 36 barrier passes instead of 78.
__global__ void __launch_bounds__(256)
ebsw_sortdist(const float* __restrict__ xp, const float* __restrict__ yp,
              float* __restrict__ dist, int l0, int Lcur) {
    __shared__ float sx[NPAD];
    __shared__ float sy[NPAD];
    __shared__ float red[256];

    const int tid = threadIdx.x;
    const int bl  = blockIdx.x;          // b * Lcur + lc
    const int b   = bl / Lcur;
    const int lc  = bl % Lcur;
    const size_t gbase = (size_t)bl * N_;

    // Async global -> LDS staging (ASYNCcnt path, bypasses VGPRs).
    // Padded layout keeps each aligned 4-float group contiguous, so B128 is legal.
    for (int i = tid * 4; i < N_; i += 256 * 4) {
        uint32_t lx = (uint32_t)(uintptr_t)&sx[PADF(i)];
        uint32_t ly = (uint32_t)(uintptr_t)&sy[PADF(i)];
        const float* gx = xp + gbase + i;
        const float* gy = yp + gbase + i;
        asm volatile("global_load_async_to_lds_b128 %0, %1, off"
                     :: "v"(lx), "v"(gx) : "memory");
        asm volatile("global_load_async_to_lds_b128 %0, %1, off"
                     :: "v"(ly), "v"(gy) : "memory");
    }
    asm volatile("s_wait_asynccnt 0x0" ::: "memory");
    __syncthreads();

    float rx[16], ry[16];
    const int base = tid * 16;           // logical chunk base
    const int pb   = tid * 20;           // padded chunk base (80B aligned -> b128 ok)

    // Stages k = 2..16: entirely in registers (10 passes, zero barriers)
#pragma unroll
    for (int v = 0; v < 16; ++v) { rx[v] = sx[pb + v]; ry[v] = sy[pb + v]; }
    reg_tail(rx, ry, base, 2u, 1);
    reg_tail(rx, ry, base, 4u, 2);
    reg_tail(rx, ry, base, 8u, 4);
    reg_tail(rx, ry, base, 16u, 8);
#pragma unroll
    for (int v = 0; v < 16; ++v) { sx[pb + v] = rx[v]; sy[pb + v] = ry[v]; }
    __syncthreads();

    // Stages k = 32..4096: LDS passes for j>=16, register tail for j=8..1
    for (unsigned k = 32; k <= (unsigned)N_; k <<= 1) {
        for (unsigned j = k >> 1; j >= 16; j >>= 1) {
            for (unsigned i = tid; i < (unsigned)N_; i += 256) {
                unsigned ixj = i ^ j;
                if (ixj > i) {
                    const bool up = ((i & k) == 0);
                    const int pi = PADF((int)i), pj = PADF((int)ixj);
                    float a0 = sx[pi], a1 = sx[pj];
                    if ((a0 > a1) == up) { sx[pi] = a1; sx[pj] = a0; }
                    float b0 = sy[pi], b1 = sy[pj];
                    if ((b0 > b1) == up) { sy[pi] = b1; sy[pj] = b0; }
                }
            }
            __syncthreads();
        }
#pragma unroll
        for (int v = 0; v < 16; ++v) { rx[v] = sx[pb + v]; ry[v] = sy[pb + v]; }
        reg_tail(rx, ry, base, k, 8);
        if (k < (unsigned)N_) {
#pragma unroll
            for (int v = 0; v < 16; ++v) { sx[pb + v] = rx[v]; sy[pb + v] = ry[v]; }
            __syncthreads();
        }
    }

    // Reduction straight from registers (sorted chunks live in rx/ry)
    float acc = 0.0f;
#pragma unroll
    for (int v = 0; v < 16; ++v) {
        float d = rx[v] - ry[v];
        acc += d * d;
    }
    red[tid] = acc;
    __syncthreads();
    for (int s = 128; s > 0; s >>= 1) {
        if (tid < s) red[tid] += red[tid + s];
        __syncthreads();
    }
    if (tid == 0) dist[(size_t)b * L_ + l0 + lc] = red[0];
}

// ---------------- IMH accept/reject chain + final loss --------------------------------
__global__ void ebsw_chain(const float* __restrict__ dist, const float* __restrict__ logu,
                           float* __restrict__ out) {
    __shared__ float part[B_];
    const int b = threadIdx.x;
    if (b < B_) {
        const float* db = dist + (size_t)b * L_;
        float cur = db[0];
        float acc = cur;
        for (int l = 1; l < L_; ++l) {
            float dn = db[l];
            float ar = fminf(0.0f, dn - cur);
            if (logu[(size_t)b * L_ + (l - 1)] <= ar) cur = dn;
            acc += cur;
        }
        part[b] = sqrtf(acc / (float)L_);   // P = 2 -> ^(1/2)
    }
    __syncthreads();
    if (b == 0) {
        float s = 0.0f;
        for (int i = 0; i < B_; ++i) s += part[i];
        out[0] = s / (float)B_;
    }
}

// ---------------- Host launcher --------------------------------------------------------
extern "C" void kernel_launch(void* const* d_in, const int* in_sizes, int n_in,
                              void* d_out, int out_size, void* d_ws, size_t ws_size,
                              hipStream_t stream) {
    const float* x = (const float*)d_in[0];
    const float* y = (const float*)d_in[1];
    float* out = (float*)d_out;

    float* theta4 = (float*)d_ws;                       // B*L*4
    float* logu   = theta4 + (size_t)B_ * L_ * 4;       // B*L
    float* dist   = logu   + (size_t)B_ * L_;           // B*L
    float* stage  = dist   + (size_t)B_ * L_;           // xp / yp staging

    const size_t fixedBytes = (size_t)B_ * L_ * 6 * sizeof(float);
    const size_t avail = (ws_size > fixedBytes) ? (ws_size - fixedBytes) : 0;
    const size_t perL  = (size_t)2 * B_ * N_ * sizeof(float);   // xp+yp per direction
    long lcMax = (long)(avail / perL);
    int Lc = (int)((lcMax / 16) * 16);
    if (Lc < 16) Lc = 16;
    if (Lc > L_) Lc = L_;

    ebsw_rng<<<(B_ * L_ + 255) / 256, 256, 0, stream>>>(theta4, logu);

    for (int l0 = 0; l0 < L_; l0 += Lc) {
        int cur = (L_ - l0 < Lc) ? (L_ - l0) : Lc;      // multiple of 16 (L=128)
        float* xps = stage;
        float* yps = stage + (size_t)B_ * cur * N_;
        dim3 g(N_ / 128, cur / 16, B_);
        ebsw_proj_wmma<<<g, 256, 0, stream>>>(x, y, theta4, xps, yps, l0, cur);
        ebsw_sortdist<<<B_ * cur, 256, 0, stream>>>(xps, yps, dist, l0, cur);
    }

    ebsw_chain<<<1, B_, 0, stream>>>(dist, logu, out);
}